// GraphSAGENet_11742440587922
// MI455X (gfx1250) — compile-verified
//
#include <hip/hip_runtime.h>

// GraphSAGE (2-layer, mean aggr) x 10 window steps + linear head, for MI455X (gfx1250).
// f32 end-to-end; layer-2 GEMMs use V_WMMA_F32_16X16X4_F32 (wave32 WMMA).
//
// Pipeline (all on `stream`, graph-capture safe, deterministic work per call):
//  1) deg count (atomics)        -> deg[N]
//  2) block scan                 -> row_start[N+1], cursor[N], inv_deg[N]
//  3) edge scatter               -> sorted_src[E]  (CSR by dst)
//  4) agg1: per-node scalar mean over edges for all 10 t
//  5) layer1 elementwise (rank-1)-> h1[t][n][64]
//  6) agg2: wave per (n,t), 2 feats/lane, 256B/row gather (L2-resident planes)
//  7) out init with blin
//  8) layer2 WMMA + fused relu + Wlin reduction -> atomicAdd into out

typedef __attribute__((ext_vector_type(2))) float v2f;
typedef __attribute__((ext_vector_type(8))) float v8f;

#define WINDOW 10
#define HIDDEN 64

// ---------------------------------------------------------------- utilities
__global__ void k_zero_i32(int* p, int n) {
    int i = blockIdx.x * blockDim.x + threadIdx.x;
    if (i < n) p[i] = 0;
}

__global__ void k_init_out(float* out, const float* blin, int n) {
    int i = blockIdx.x * blockDim.x + threadIdx.x;
    if (i < n) out[i] = blin[0];
}

// ---------------------------------------------------------------- CSR build
__global__ void k_count(const int* __restrict__ ei, int E, int* __restrict__ deg) {
    int e = blockIdx.x * blockDim.x + threadIdx.x;
    if (e < E) atomicAdd(&deg[ei[E + e]], 1);  // dst = edge_index[1][e]
}

// Single-block exclusive scan over deg -> row_start/cursor, plus inv_deg.
__global__ __launch_bounds__(1024) void k_scan_build(const int* __restrict__ deg, int n,
                                                     int* __restrict__ row_start,
                                                     int* __restrict__ cursor,
                                                     float* __restrict__ inv_deg) {
    __shared__ int sdata[1024];
    __shared__ int s_running;
    int tid = threadIdx.x;
    if (tid == 0) s_running = 0;
    __syncthreads();
    for (int base = 0; base < n; base += 1024) {
        int i = base + tid;
        int v = (i < n) ? deg[i] : 0;
        sdata[tid] = v;
        __syncthreads();
        // Hillis-Steele inclusive scan
        for (int off = 1; off < 1024; off <<= 1) {
            int t2 = (tid >= off) ? sdata[tid - off] : 0;
            __syncthreads();
            sdata[tid] += t2;
            __syncthreads();
        }
        int incl = sdata[tid];
        int excl = incl - v;
        if (i < n) {
            int rs = s_running + excl;
            row_start[i] = rs;
            cursor[i]    = rs;
            inv_deg[i]   = 1.0f / fmaxf((float)v, 1.0f);
        }
        __syncthreads();
        if (tid == 1023) s_running += incl;
        __syncthreads();
    }
    if (tid == 0) row_start[n] = s_running;
}

__global__ void k_scatter(const int* __restrict__ ei, int E,
                          int* __restrict__ cursor, int* __restrict__ sorted_src) {
    int e = blockIdx.x * blockDim.x + threadIdx.x;
    if (e < E) {
        int s = ei[e];
        int d = ei[E + e];
        int pos = atomicAdd(&cursor[d], 1);
        sorted_src[pos] = s;
    }
}

// ---------------------------------------------------------------- layer 1
// agg1[t][n] = mean_{j in N(n)} x[j][t]   (scalar features, all 10 t at once)
__global__ void k_agg1(const float* __restrict__ x, const int* __restrict__ row_start,
                       const int* __restrict__ sorted_src, const float* __restrict__ inv_deg,
                       int N, float* __restrict__ agg1) {
    int n = blockIdx.x * blockDim.x + threadIdx.x;
    if (n >= N) return;
    float acc[WINDOW];
#pragma unroll
    for (int w = 0; w < WINDOW; ++w) acc[w] = 0.0f;
    int e0 = row_start[n], e1 = row_start[n + 1];
    for (int e = e0; e < e1; ++e) {
        const float* xs = x + (size_t)sorted_src[e] * WINDOW;
#pragma unroll
        for (int w = 0; w < WINDOW; ++w) acc[w] += xs[w];
    }
    float idg = inv_deg[n];
#pragma unroll
    for (int w = 0; w < WINDOW; ++w) agg1[(size_t)w * N + n] = acc[w] * idg;
}

// h1[t][n][f] = relu(agg1[t][n]*W1l[f] + b1[f] + x[n][t]*W1r[f])   (rank-1, no GEMM)
__global__ void k_layer1(const float* __restrict__ x, const float* __restrict__ agg1,
                         const float* __restrict__ W1l, const float* __restrict__ b1,
                         const float* __restrict__ W1r, int N, float* __restrict__ h1) {
    long idx = (long)blockIdx.x * blockDim.x + threadIdx.x;
    long total = (long)N * WINDOW * HIDDEN;
    if (idx >= total) return;
    int f  = (int)(idx & (HIDDEN - 1));
    long nt = idx >> 6;               // nt = t*N + n
    int n  = (int)(nt % N);
    int t  = (int)(nt / N);
    float a  = agg1[(size_t)t * N + n];
    float xv = x[(size_t)n * WINDOW + t];
    float v  = a * W1l[f] + b1[f] + xv * W1r[f];
    h1[idx] = fmaxf(v, 0.0f);
}

// ---------------------------------------------------------------- layer-2 aggregation
// One wave32 per (node, t); lane owns 2 contiguous features (float2).
// Each gathered h1 row = 256B contiguous (full cacheline); planes are L2-resident.
__global__ __launch_bounds__(128) void k_agg2(const float* __restrict__ h1,
                                              const int* __restrict__ row_start,
                                              const int* __restrict__ sorted_src,
                                              const float* __restrict__ inv_deg,
                                              int N, float* __restrict__ agg2) {
    int wid  = threadIdx.x >> 5;
    int lane = threadIdx.x & 31;
    long task = (long)blockIdx.x * 4 + wid;
    if (task >= (long)N * WINDOW) return;
    int n = (int)(task % N);   // node fastest -> neighboring waves share edge-list lines
    int t = (int)(task / N);
    const float* hb = h1 + (size_t)t * N * HIDDEN;
    int e0 = row_start[n], e1 = row_start[n + 1];
    v2f acc; acc.x = 0.0f; acc.y = 0.0f;
    for (int e = e0; e < e1; ++e) {
        int s = sorted_src[e];
        if (e + 1 < e1)  // hide gather latency -> global_prefetch_b8
            __builtin_prefetch(hb + (size_t)sorted_src[e + 1] * HIDDEN + lane * 2, 0, 0);
        v2f v = *(const v2f*)(hb + (size_t)s * HIDDEN + lane * 2);
        acc += v;
    }
    float idg = inv_deg[n];
    acc.x *= idg; acc.y *= idg;
    *(v2f*)(agg2 + ((size_t)t * N + n) * HIDDEN + lane * 2) = acc;
}

// ---------------------------------------------------------------- layer-2 WMMA + head
// One wave per (16-node tile, t). D(16x64) = relu(AGG(16x64)@W2l + H1(16x64)@W2r + b2),
// then fused head: out[n] += sum_f D[n][f] * Wlin[t*64+f].
// V_WMMA_F32_16X16X4_F32 layouts (wave32):
//   A 16x4 : lane m=lane&15, VGPR pair holds K = (lane>>4)*2 + {0,1}
//   B 4x16 : lane n=lane&15, VGPR pair holds K = (lane>>4)*2 + {0,1}
//   C 16x16: component r = row (r + 8*(lane>=16)), col = lane&15
__global__ __launch_bounds__(32) void k_layer2_wmma(const float* __restrict__ agg2,
                                                    const float* __restrict__ h1,
                                                    const float* __restrict__ W2l,
                                                    const float* __restrict__ W2r,
                                                    const float* __restrict__ b2,
                                                    const float* __restrict__ wlin,
                                                    float* __restrict__ out, int N) {
    const int tile = blockIdx.x;       // 16-node tile
    const int t    = blockIdx.y;       // window step
    const int lane = threadIdx.x;      // wave32
    const int m    = lane & 15;
    const int koff = (lane >> 4) << 1; // {0,2}

    int node = tile * 16 + m;
    int node_c = (node < N) ? node : (N - 1);       // clamp (EXEC must stay full for WMMA)
    const float* aggRow = agg2 + ((size_t)t * N + node_c) * HIDDEN;
    const float* hRow   = h1   + ((size_t)t * N + node_c) * HIDDEN;

    v8f acc[4];
#pragma unroll
    for (int ct = 0; ct < 4; ++ct) {
        float bv = b2[ct * 16 + m];   // bias is per output column
#pragma unroll
        for (int r = 0; r < 8; ++r) acc[ct][r] = bv;
    }

#pragma unroll
    for (int k = 0; k < 16; ++k) {           // K = 64 in steps of 4
        int c0 = 4 * k + koff;
        v2f a1 = *(const v2f*)(aggRow + c0); // aggregate branch
        v2f a2 = *(const v2f*)(hRow + c0);   // root (self) branch
#pragma unroll
        for (int ct = 0; ct < 4; ++ct) {
            int col = ct * 16 + m;
            v2f bl, br;
            bl.x = W2l[(size_t)c0 * HIDDEN + col];
            bl.y = W2l[(size_t)(c0 + 1) * HIDDEN + col];
            br.x = W2r[(size_t)c0 * HIDDEN + col];
            br.y = W2r[(size_t)(c0 + 1) * HIDDEN + col];
            acc[ct] = __builtin_amdgcn_wmma_f32_16x16x4_f32(
                false, a1, false, bl, (short)0, acc[ct], false, false);
            acc[ct] = __builtin_amdgcn_wmma_f32_16x16x4_f32(
                false, a2, false, br, (short)0, acc[ct], false, false);
        }
    }

    // relu + per-lane partial of the Wlin dot (column = ct*16 + m)
    float s[8];
#pragma unroll
    for (int r = 0; r < 8; ++r) s[r] = 0.0f;
#pragma unroll
    for (int ct = 0; ct < 4; ++ct) {
        float wl = wlin[t * HIDDEN + ct * 16 + m];
#pragma unroll
        for (int r = 0; r < 8; ++r) s[r] += fmaxf(acc[ct][r], 0.0f) * wl;
    }
    // reduce across the 16 lanes of each half (rows 0-7 live in lanes 0-15, 8-15 in 16-31)
#pragma unroll
    for (int mask = 1; mask < 16; mask <<= 1) {
#pragma unroll
        for (int r = 0; r < 8; ++r) s[r] += __shfl_xor(s[r], mask, 32);
    }
    if (m == 0) {
        int rbase = tile * 16 + ((lane >> 4) * 8);
#pragma unroll
        for (int r = 0; r < 8; ++r)
            if (rbase + r < N) atomicAdd(&out[rbase + r], s[r]);  // accumulate over t
    }
}

// ---------------------------------------------------------------- launcher
extern "C" void kernel_launch(void* const* d_in, const int* in_sizes, int n_in,
                              void* d_out, int out_size, void* d_ws, size_t ws_size,
                              hipStream_t stream) {
    const float* x    = (const float*)d_in[0];
    const int*   ei   = (const int*)d_in[1];   // edge_index, int32 on device (JAX default)
    const float* W1l  = (const float*)d_in[2];
    const float* b1   = (const float*)d_in[3];
    const float* W1r  = (const float*)d_in[4];
    const float* W2l  = (const float*)d_in[5];
    const float* b2   = (const float*)d_in[6];
    const float* W2r  = (const float*)d_in[7];
    const float* Wlin = (const float*)d_in[8];
    const float* blin = (const float*)d_in[9];
    float* out = (float*)d_out;

    const int N = in_sizes[0] / WINDOW;  // 50000
    const int E = in_sizes[1] / 2;       // 1600000

    // Workspace carve-out (~266 MB total):
    char* ws = (char*)d_ws;
    size_t off = 0;
    auto take = [&](size_t bytes) -> char* {
        char* p = ws + off;
        off = (off + bytes + 255) & ~(size_t)255;
        return p;
    };
    int*   deg       = (int*)  take((size_t)N * 4);
    int*   row_start = (int*)  take((size_t)(N + 1) * 4);
    int*   cursor    = (int*)  take((size_t)N * 4);
    float* inv_deg   = (float*)take((size_t)N * 4);
    int*   sorted    = (int*)  take((size_t)E * 4);
    float* agg1      = (float*)take((size_t)WINDOW * N * 4);
    float* h1        = (float*)take((size_t)WINDOW * N * HIDDEN * 4);
    float* agg2      = (float*)take((size_t)WINDOW * N * HIDDEN * 4);
    (void)ws_size; (void)n_in; (void)out_size;

    const int TB = 256;
    k_zero_i32  <<<(N + TB - 1) / TB, TB, 0, stream>>>(deg, N);
    k_count     <<<(E + TB - 1) / TB, TB, 0, stream>>>(ei, E, deg);
    k_scan_build<<<1, 1024, 0, stream>>>(deg, N, row_start, cursor, inv_deg);
    k_scatter   <<<(E + TB - 1) / TB, TB, 0, stream>>>(ei, E, cursor, sorted);
    k_agg1      <<<(N + TB - 1) / TB, TB, 0, stream>>>(x, row_start, sorted, inv_deg, N, agg1);

    long tot1 = (long)N * WINDOW * HIDDEN;
    k_layer1    <<<(unsigned)((tot1 + TB - 1) / TB), TB, 0, stream>>>(x, agg1, W1l, b1, W1r, N, h1);

    long tasks = (long)N * WINDOW;               // wave per (node, t)
    k_agg2      <<<(unsigned)((tasks + 3) / 4), 128, 0, stream>>>(h1, row_start, sorted, inv_deg, N, agg2);

    k_init_out  <<<(N + TB - 1) / TB, TB, 0, stream>>>(out, blin, N);
    dim3 g2((N + 15) / 16, WINDOW);
    k_layer2_wmma<<<g2, 32, 0, stream>>>(agg2, h1, W2l, W2r, b2, Wlin, out, N);
}